// GCNNet_32908039422339
// MI455X (gfx1250) — compile-verified
//
#include <hip/hip_runtime.h>

// ---------------------------------------------------------------------------
// GCN (2x GCNConv + linear head) for MI455X / gfx1250.
// fp32 WMMA GEMMs fed by Tensor-Data-Mover (TDM) tiles double-buffered in LDS.
// ---------------------------------------------------------------------------

typedef __attribute__((ext_vector_type(2))) float v2f;
typedef __attribute__((ext_vector_type(8))) float v8f;
typedef unsigned int u32x4 __attribute__((ext_vector_type(4)));
typedef int i32x8 __attribute__((ext_vector_type(8)));
typedef int i32x4 __attribute__((ext_vector_type(4)));

#define FIN   500
#define HID1  128
#define HID2  64
#define NCLS  10

__device__ __forceinline__ v8f wmma_f32(v2f a, v2f b, v8f c) {
  return __builtin_amdgcn_wmma_f32_16x16x4_f32(
      false, a, false, b, (short)0, c, false, false);
}

// LDS byte-offset of a __shared__ object (generic -> AS3 -> integer).
#define LDS_OFF(p) \
  ((unsigned)(unsigned long long)(__attribute__((address_space(3))) const void*)(p))

// ---- TDM: issue a 2D tile load Global->LDS (data_size = 4B) ---------------
// tensor_d0/d1: tensor extents for OOB zero-fill; tile_d0/d1: tile shape;
// stride0: elements between rows; pad_i/pad_a: LDS padding (ISA encodings).
__device__ __forceinline__ void tdm_load_2d(
    unsigned lds_off, const void* gptr,
    unsigned tensor_d0, unsigned tensor_d1,
    unsigned tile_d0, unsigned tile_d1,
    unsigned long long stride0, int pad_i, int pad_a, int pad_en) {
  unsigned long long ga = (unsigned long long)(size_t)gptr;

  // group0: count=1 | lds_addr<<32 ; global_addr[56:0] | type=2 @ [127:126]
  unsigned long long g0q0 = 1ull | ((unsigned long long)lds_off << 32);
  unsigned long long g0q1 = (ga & ((1ull << 57) - 1ull)) | (2ull << 62);
  u32x4 g0 = {(unsigned)g0q0, (unsigned)(g0q0 >> 32),
              (unsigned)g0q1, (unsigned)(g0q1 >> 32)};

  // group1 (256b): mask=0 | data_size=2(4B)@16 | pad_en@20 | pad_i@24:22 |
  //   pad_a@31:25 | abar=0 | tensor_d0@79:48 | tensor_d1@111:80 |
  //   tile_d0@127:112 | tile_d1@143:128 | tile_d2=0@159:144 |
  //   stride0@207:160 | stride1=0@255:208
  unsigned long long q0 = (2ull << 16)
                        | ((unsigned long long)(pad_en & 1) << 20)
                        | ((unsigned long long)(pad_i & 7) << 22)
                        | ((unsigned long long)(pad_a & 127) << 25)
                        | ((unsigned long long)(tensor_d0 & 0xFFFFu) << 48);
  unsigned long long q1 = (unsigned long long)(tensor_d0 >> 16)
                        | ((unsigned long long)tensor_d1 << 16)
                        | ((unsigned long long)(tile_d0 & 0xFFFFu) << 48);
  unsigned long long q2 = (unsigned long long)(tile_d1 & 0xFFFFu)
                        | ((stride0 & 0xFFFFFFFFull) << 32);
  unsigned long long q3 = (stride0 >> 32) & 0xFFFFull;
  i32x8 g1 = {(int)q0, (int)(q0 >> 32), (int)q1, (int)(q1 >> 32),
              (int)q2, (int)(q2 >> 32), (int)q3, (int)(q3 >> 32)};
  i32x4 g2 = {1, 1, 0, 0};  // tensor_dim2=1, tensor_dim3=1 (unused: tile_d2=0)
  i32x4 g3 = {0, 0, 0, 0};
  i32x8 gx = {0, 0, 0, 0, 0, 0, 0, 0};  // extra group (6-arg clang-23 form)
  __builtin_amdgcn_tensor_load_to_lds(g0, g1, g2, g3, gx, 0);
}

// ---------------- degree / normalization ----------------------------------

__global__ void k_fill1(float* __restrict__ p, int n) {
  int i = blockIdx.x * blockDim.x + threadIdx.x;
  if (i < n) p[i] = 1.0f;
}

__global__ void k_degcount(const long long* __restrict__ dst,
                           float* __restrict__ deg, int E) {
  int i = blockIdx.x * blockDim.x + threadIdx.x;
  if (i < E) unsafeAtomicAdd(&deg[(int)dst[i]], 1.0f);
}

__global__ void k_rsqrt(float* __restrict__ p, int n) {
  int i = blockIdx.x * blockDim.x + threadIdx.x;
  if (i < n) p[i] = rsqrtf(p[i]);
}

// ---------------- TDM + WMMA GEMM: C[N x OUT] = A[N x K] * B[K x OUT] ------
// 4 waves/WG, 128 rows/WG, 2 row-tiles (32 rows) per wave, all NT col tiles.
// K consumed in KC=32 chunks, double-buffered in LDS via TENSOR_LOAD_TO_LDS.
// OOB K/N tails zero-filled by the TDM (tensor_dim), zeros are benign in GEMM.

template <int NT>
__global__ __launch_bounds__(128) void k_gemm_wmma(
    const float* __restrict__ A, const float* __restrict__ B,
    float* __restrict__ C, int N, int K) {
  constexpr int OUT  = NT * 16;
  constexpr int KC   = 32;        // K-chunk
  constexpr int ASTR = KC + 2;    // A row stride in LDS (pad 2 -> 8B aligned)
  constexpr int BSTR = OUT + 1;   // B row stride in LDS (pad 1)
  constexpr int BPADI = (OUT == 128) ? 6 : 5;  // pad interval: 128 / 64 dwords

  __shared__ float As[2][128 * ASTR + 4];
  __shared__ float Bs[2][KC * BSTR + 4];

  const int lane = threadIdx.x & 31;
  const int wave = threadIdx.x >> 5;
  const int half = lane >> 4;
  const int l16  = lane & 15;
  const int ko   = half * 2;
  const int wrow = wave * 32;                 // wave's local row base
  const int row0b = blockIdx.x * 128;         // workgroup's global row base

  v8f acc[2][NT];
#pragma unroll
  for (int r = 0; r < 2; ++r)
#pragma unroll
    for (int t = 0; t < NT; ++t) {
      v8f z = {0.f, 0.f, 0.f, 0.f, 0.f, 0.f, 0.f, 0.f};
      acc[r][t] = z;
    }

  const int nch = (K + KC - 1) / KC;

  // Chunk c: A tile = rows[row0b..+128) x K[kk..kk+32)  (row stride K)
  //          B tile = K[kk..kk+32) x OUT                 (row stride OUT)
  auto issue = [&](int c, int buf) {
    const int kk = c * KC;
    tdm_load_2d(LDS_OFF(&As[buf][0]),
                A + (size_t)row0b * K + kk,
                (unsigned)(K - kk), (unsigned)(N - row0b),
                KC, 128, (unsigned long long)K,
                /*pad_i 32dw=*/4, /*pad_a 2dw=*/1, 1);
    tdm_load_2d(LDS_OFF(&Bs[buf][0]),
                B + (size_t)kk * OUT,
                (unsigned)OUT, (unsigned)(K - kk),
                OUT, KC, (unsigned long long)OUT,
                BPADI, /*pad_a 1dw=*/0, 1);
  };

  if (threadIdx.x < 32) issue(0, 0);

  for (int c = 0; c < nch; ++c) {
    if (threadIdx.x < 32) {
      if (c + 1 < nch) {
        issue(c + 1, (c + 1) & 1);
        __builtin_amdgcn_s_wait_tensorcnt(2);  // chunk c's 2 descriptors done
      } else {
        __builtin_amdgcn_s_wait_tensorcnt(0);
      }
    }
    __syncthreads();

    const float* Ab = As[c & 1];
    const float* Bb = Bs[c & 1];
#pragma unroll
    for (int kl = 0; kl < KC; kl += 4) {
      v2f a0 = *(const v2f*)(Ab + (wrow + l16) * ASTR + kl + ko);
      v2f a1 = *(const v2f*)(Ab + (wrow + 16 + l16) * ASTR + kl + ko);
#pragma unroll
      for (int t = 0; t < NT; ++t) {
        v2f b;
        b.x = Bb[(kl + ko) * BSTR + t * 16 + l16];
        b.y = Bb[(kl + ko + 1) * BSTR + t * 16 + l16];
        acc[0][t] = wmma_f32(a0, b, acc[0][t]);
        acc[1][t] = wmma_f32(a1, b, acc[1][t]);
      }
    }
    __syncthreads();
  }

  // D layout: VGPR g -> row (g + half*8), col l16
#pragma unroll
  for (int r = 0; r < 2; ++r)
#pragma unroll
    for (int t = 0; t < NT; ++t)
#pragma unroll
      for (int g = 0; g < 8; ++g) {
        int row = row0b + wrow + r * 16 + half * 8 + g;
        if (row < N) C[(size_t)row * OUT + t * 16 + l16] = acc[r][t][g];
      }
}

// ---------------- self-loop init: agg = h * dis^2 --------------------------

template <int F>
__global__ void k_selfscale(const float* __restrict__ h,
                            const float* __restrict__ dis,
                            float* __restrict__ agg, int n) {
  int i = blockIdx.x * blockDim.x + threadIdx.x;
  if (i < n * F) {
    float d = dis[i / F];
    agg[i] = h[i] * d * d;
  }
}

// ---------------- edge aggregation: agg[dst] += norm * h[src] --------------

template <int F>
__global__ void k_edge_agg(const long long* __restrict__ src,
                           const long long* __restrict__ dst,
                           const float* __restrict__ dis,
                           const float* __restrict__ h,
                           float* __restrict__ agg, int E) {
  constexpr int VPT = F / 32;
  int gtid = blockIdx.x * blockDim.x + threadIdx.x;
  int wave = gtid >> 5;
  int lane = threadIdx.x & 31;
  int nwaves = (gridDim.x * blockDim.x) >> 5;
  for (int e = wave; e < E; e += nwaves) {
    int s = (int)src[e];
    int d = (int)dst[e];
    float norm = dis[s] * dis[d];
    const float* hp = h + (size_t)s * F + lane * VPT;
    float* ap = agg + (size_t)d * F + lane * VPT;
    float v[VPT];
#pragma unroll
    for (int j = 0; j < VPT; ++j) v[j] = hp[j];
#pragma unroll
    for (int j = 0; j < VPT; ++j) unsafeAtomicAdd(ap + j, norm * v[j]);
  }
}

// ---------------- bias + relu (in place) -----------------------------------

template <int F>
__global__ void k_bias_relu(float* __restrict__ x, const float* __restrict__ b,
                            int n) {
  int i = blockIdx.x * blockDim.x + threadIdx.x;
  if (i < n * F) {
    float v = x[i] + b[i & (F - 1)];
    x[i] = v > 0.f ? v : 0.f;
  }
}

// ---------------- classifier: out = a @ Wc + bc ----------------------------

__global__ void k_classifier(const float* __restrict__ a,
                             const float* __restrict__ Wc,
                             const float* __restrict__ bc,
                             float* __restrict__ out, int n) {
  int i = blockIdx.x * blockDim.x + threadIdx.x;
  if (i >= n) return;
  float row[HID2];
#pragma unroll
  for (int k = 0; k < HID2; ++k) row[k] = a[(size_t)i * HID2 + k];
#pragma unroll
  for (int c = 0; c < NCLS; ++c) {
    float s = bc[c];
#pragma unroll
    for (int k = 0; k < HID2; ++k) s += row[k] * Wc[k * NCLS + c];
    out[(size_t)i * NCLS + c] = s;
  }
}

// ---------------------------------------------------------------------------

extern "C" void kernel_launch(void* const* d_in, const int* in_sizes, int n_in,
                              void* d_out, int out_size, void* d_ws,
                              size_t ws_size, hipStream_t stream) {
  (void)n_in; (void)out_size; (void)ws_size;
  const float*     x   = (const float*)d_in[0];
  const long long* ei  = (const long long*)d_in[1];
  const float*     W1  = (const float*)d_in[2];
  const float*     b1  = (const float*)d_in[3];
  const float*     W2  = (const float*)d_in[4];
  const float*     b2  = (const float*)d_in[5];
  const float*     Wc  = (const float*)d_in[6];
  const float*     bc  = (const float*)d_in[7];
  float*           out = (float*)d_out;

  const int N = in_sizes[0] / FIN;
  const int E = in_sizes[1] / 2;
  const long long* srcp = ei;
  const long long* dstp = ei + E;

  // Workspace layout (floats). h2/agg2 overlay the dead h1 region.
  float* ws   = (float*)d_ws;
  float* dis  = ws;                          // N
  float* h1   = ws + N;                      // N*HID1
  float* agg1 = ws + (size_t)N * (1 + HID1); // N*HID1
  float* h2   = h1;                          // N*HID2
  float* agg2 = h1 + (size_t)N * HID2;       // N*HID2

  const int T = 256;
  int nN  = (N + T - 1) / T;
  int nE  = (E + T - 1) / T;
  int n1  = (N * HID1 + T - 1) / T;
  int n2  = (N * HID2 + T - 1) / T;
  int gWG = (N + 127) / 128;

  k_fill1   <<<nN, T, 0, stream>>>(dis, N);
  k_degcount<<<nE, T, 0, stream>>>(dstp, dis, E);
  k_rsqrt   <<<nN, T, 0, stream>>>(dis, N);

  k_gemm_wmma<8> <<<gWG, 128, 0, stream>>>(x, W1, h1, N, FIN);
  k_selfscale<HID1><<<n1, T, 0, stream>>>(h1, dis, agg1, N);
  k_edge_agg<HID1><<<4096, 256, 0, stream>>>(srcp, dstp, dis, h1, agg1, E);
  k_bias_relu<HID1><<<n1, T, 0, stream>>>(agg1, b1, N);

  k_gemm_wmma<4> <<<gWG, 128, 0, stream>>>(agg1, W2, h2, N, HID1);
  k_selfscale<HID2><<<n2, T, 0, stream>>>(h2, dis, agg2, N);
  k_edge_agg<HID2><<<4096, 256, 0, stream>>>(srcp, dstp, dis, h2, agg2, E);
  k_bias_relu<HID2><<<n2, T, 0, stream>>>(agg2, b2, N);

  k_classifier<<<nN, T, 0, stream>>>(agg2, Wc, bc, out, N);
}